// NMSHead_9156870275261
// MI455X (gfx1250) — compile-verified
//
#include <hip/hip_runtime.h>
#include <stdint.h>

// Problem constants (from reference)
#define BATCH   16
#define Hh      2048
#define Ww      2048
#define KCAP    ((Hh * Ww) / 8)   // 524288
#define MINVAL  1e-5f

typedef unsigned int v4u __attribute__((ext_vector_type(4)));
typedef int          v8i __attribute__((ext_vector_type(8)));
typedef int          v4i __attribute__((ext_vector_type(4)));
typedef float        v2f __attribute__((ext_vector_type(2)));
typedef float        v8f __attribute__((ext_vector_type(8)));

// ---------------------------------------------------------------------------
// Pass 1: per-row 5x5 max-filter mask + per-row peak count.
// One block per (image,row). TDM loads the 5-row band into LDS.
// ---------------------------------------------------------------------------
__global__ __launch_bounds__(256) void nms_mask_kernel(
    const float* __restrict__ in,
    unsigned* __restrict__ maskWords,     // [BATCH*Hh][Ww/32]
    unsigned* __restrict__ rowCounts)     // [BATCH*Hh]
{
    const int gb = blockIdx.x;            // b*Hh + r
    const int b  = gb >> 11;
    const int r  = gb & (Hh - 1);
    const int t  = threadIdx.x;

    __shared__ float    rows[5][Ww];      // 40 KiB band
    __shared__ float    vm[Ww + 4];       // vertical max + column halo
    __shared__ unsigned cnt;

    if (t == 0) cnt = 0u;
    if (t < 2) { vm[t] = 0.0f; vm[Ww + 2 + t] = 0.0f; }

    // Zero-fill band rows that fall outside the image (reference pads with 0).
    for (int dr = 0; dr < 5; ++dr) {
        int rr = r - 2 + dr;
        if (rr < 0 || rr >= Hh) {
            for (int j = t; j < Ww; j += 256) rows[dr][j] = 0.0f;
        }
    }

    // ---- Tensor Data Mover: DMA rows [clo, chi] into LDS -------------------
    {
        const int clo   = (r - 2 < 0) ? 0 : (r - 2);
        const int chi   = (r + 2 > Hh - 1) ? (Hh - 1) : (r + 2);
        const unsigned nrows = (unsigned)(chi - clo + 1);
        const unsigned lds_off = (unsigned)(uintptr_t)(&rows[clo - (r - 2)][0]);
        const unsigned long long gaddr =
            (unsigned long long)(uintptr_t)(in + ((size_t)b * Hh + (size_t)clo) * Ww);

        // D# group 0: [1:0]=count=1, [63:32]=lds_addr, [120:64]=global_addr, [127:126]=type=2
        v4u g0;
        g0[0] = 1u;
        g0[1] = lds_off;
        g0[2] = (unsigned)(gaddr & 0xFFFFFFFFull);
        g0[3] = (unsigned)((gaddr >> 32) & 0x1FFFFFFull) | (2u << 30);

        // D# group 1: data_size=2(4B); tensor_dim0=Ww; tensor_dim1=nrows;
        //             tile_dim0=Ww; tile_dim1=nrows; tensor_dim0_stride=Ww
        v8i g1;
        g1[0] = (int)(2u << 16);
        g1[1] = (int)(((unsigned)Ww & 0xFFFFu) << 16);                        // dim0 lo16 -> [31:16]
        g1[2] = (int)((((unsigned)Ww >> 16) & 0xFFFFu) | ((nrows & 0xFFFFu) << 16));
        g1[3] = (int)(((nrows >> 16) & 0xFFFFu) | (((unsigned)Ww & 0xFFFFu) << 16)); // tile_dim0
        g1[4] = (int)(nrows & 0xFFFFu);                                       // tile_dim1, tile_dim2=0
        g1[5] = (int)Ww;                                                      // stride0 lo32
        g1[6] = 0;
        g1[7] = 0;

        v4i g2 = {0, 0, 0, 0};
        v4i g3 = {0, 0, 0, 0};

        // TDM ignores EXEC: gate by a *scalar* wave-id branch so only wave 0 issues.
        if ((__builtin_amdgcn_readfirstlane((int)t) >> 5) == 0) {
#if defined(__clang_major__) && (__clang_major__ >= 23)
            v8i g4 = {0, 0, 0, 0, 0, 0, 0, 0};
            __builtin_amdgcn_tensor_load_to_lds(g0, g1, g2, g3, g4, 0);
#else
            __builtin_amdgcn_tensor_load_to_lds(g0, g1, g2, g3, 0);
#endif
            __builtin_amdgcn_s_wait_tensorcnt(0);
        }
    }
    __syncthreads();

    // Vertical 5-max into vm (with +2 column shift for halo).
    for (int j = t; j < Ww; j += 256) {
        float a = rows[0][j], bb = rows[1][j], c = rows[2][j];
        float d = rows[3][j], e = rows[4][j];
        vm[j + 2] = fmaxf(fmaxf(fmaxf(a, bb), fmaxf(c, d)), e);
    }
    __syncthreads();

    // Horizontal 5-max, peak test, ballot to 32-pixel mask words.
    unsigned localCnt = 0;
    for (int cc = 0; cc < 8; ++cc) {
        int j = cc * 256 + t;                     // lanes cover 32 contiguous pixels
        float center = rows[2][j];
        float h = fmaxf(fmaxf(vm[j], vm[j + 1]),
                        fmaxf(fmaxf(vm[j + 2], vm[j + 3]), vm[j + 4]));
        bool is_peak = (h > MINVAL) && (h == center);
        unsigned word = __builtin_amdgcn_ballot_w32(is_peak);
        if ((t & 31) == 0) {
            maskWords[(size_t)gb * (Ww / 32) + (j >> 5)] = word;
            localCnt += (unsigned)__builtin_popcount(word);
        }
    }
    if ((t & 31) == 0) atomicAdd(&cnt, localCnt);
    __syncthreads();
    if (t == 0) rowCounts[gb] = cnt;
}

// ---------------------------------------------------------------------------
// Pass 2: per-image exclusive scan of 2048 row counts. One block per image.
// The 256 thread-sums are scanned with V_WMMA_F32_16X16X4_F32: the 16x16
// matrix of sums (column = 16-element segment) is multiplied by a lower-
// triangular ones matrix L via 4 accumulating WMMAs, giving all segment-wise
// inclusive scans in one matrix op. Counts < 2^24 so f32 WMMA is exact.
// ---------------------------------------------------------------------------
__global__ __launch_bounds__(256) void scan_kernel(
    const unsigned* __restrict__ rowCounts,
    unsigned* __restrict__ rowOffsets)
{
    const int b = blockIdx.x;
    const int t = threadIdx.x;
    __shared__ float sums[256];     // per-thread sums  (X, column-major segments)
    __shared__ float scanned[256];  // segment-wise inclusive scans (Y = L*X)

    const unsigned* rc = rowCounts + (size_t)b * Hh;
    unsigned*       ro = rowOffsets + (size_t)b * Hh;

    unsigned v[8];
    unsigned run = 0;
    for (int i = 0; i < 8; ++i) {
        unsigned x = rc[t * 8 + i];
        v[i] = run;                 // local exclusive prefix
        run += x;
    }
    sums[t] = (float)run;
    __syncthreads();

    // Wave 0 (EXEC all-ones, uniform branch) performs the matrix scan.
    if (t < 32) {
        const int m  = t & 15;      // A-row / B-column / D-column index
        const int hi = t >> 4;      // lane half
        v8f acc = {};
        #pragma unroll
        for (int j = 0; j < 4; ++j) {
            // A = L[:, 4j..4j+3]: lane half selects K pair {0,1} or {2,3}.
            v2f a, bvec;
            int k0 = 4 * j + 2 * hi;
            int k1 = k0 + 1;
            a.x = (k0 <= m) ? 1.0f : 0.0f;
            a.y = (k1 <= m) ? 1.0f : 0.0f;
            // B = X[4j..4j+3, :]: vgpr0 rows {4j+hi}, vgpr1 rows {4j+2+hi}.
            bvec.x = sums[m * 16 + 4 * j + hi];
            bvec.y = sums[m * 16 + 4 * j + 2 + hi];
            acc = __builtin_amdgcn_wmma_f32_16x16x4_f32(
                false, a, false, bvec, (short)0, acc, false, false);
        }
        // D layout: vgpr v, lane -> element (M = v + 8*hi, N = m).
        #pragma unroll
        for (int vv = 0; vv < 8; ++vv) {
            scanned[m * 16 + vv + 8 * hi] = acc[vv];   // scanned[segment*16 + pos]
        }
    }
    __syncthreads();

    // Exclusive base for thread t = (sum of totals of earlier segments)
    //                              + (inclusive scan within segment) - own sum.
    const int seg = t >> 4;
    float segoff = 0.0f;
    for (int c = 0; c < seg; ++c) segoff += scanned[c * 16 + 15];
    float basef = segoff + scanned[t] - (float)run;
    unsigned base = (unsigned)(basef + 0.5f);   // exact integer in f32

    for (int i = 0; i < 8; ++i) ro[t * 8 + i] = base + v[i];
}

// ---------------------------------------------------------------------------
// Pass 3a: constant background: rows 0/1 -> 0, row 2 -> 1 for all K columns.
// ---------------------------------------------------------------------------
__global__ __launch_bounds__(256) void fill_kernel(float* __restrict__ out)
{
    size_t idx = (size_t)blockIdx.x * 256u + threadIdx.x;   // one float4 each
    size_t plane = (idx * 4u) / (size_t)KCAP;               // b*3 + row
    float val = ((plane % 3u) == 2u) ? 1.0f : 0.0f;
    float4* o = (float4*)out;
    o[idx] = make_float4(val, val, val, val);
}

// ---------------------------------------------------------------------------
// Pass 3b: deterministic rank-based scatter of peak (x, y) coordinates.
// One block (64 threads) per row; thread t owns mask word t (32 pixels).
// ---------------------------------------------------------------------------
__global__ __launch_bounds__(64) void scatter_kernel(
    const unsigned* __restrict__ maskWords,
    const unsigned* __restrict__ rowOffsets,
    float* __restrict__ out)
{
    const int gb = blockIdx.x;
    const int b  = gb >> 11;
    const int r  = gb & (Hh - 1);
    const int t  = threadIdx.x;   // 0..63

    __shared__ unsigned pc[64];
    unsigned word = maskWords[(size_t)gb * (Ww / 32) + t];
    pc[t] = (unsigned)__builtin_popcount(word);
    __syncthreads();
    for (int off = 1; off < 64; off <<= 1) {
        unsigned x = (t >= off) ? pc[t - off] : 0u;
        __syncthreads();
        pc[t] += x;
        __syncthreads();
    }
    unsigned base = rowOffsets[gb] + ((t == 0) ? 0u : pc[t - 1]);

    float* xrow = out + (size_t)b * 3u * KCAP;
    float* yrow = xrow + KCAP;
    const float fy = (float)r;

    unsigned w = word;
    unsigned k = 0;
    while (w) {
        int bit = __builtin_ctz(w);
        w &= w - 1u;
        unsigned idx = base + k; ++k;
        if (idx < (unsigned)KCAP) {
            xrow[idx] = (float)(t * 32 + bit);
            yrow[idx] = fy;
        }
    }
}

// ---------------------------------------------------------------------------
extern "C" void kernel_launch(void* const* d_in, const int* in_sizes, int n_in,
                              void* d_out, int out_size, void* d_ws, size_t ws_size,
                              hipStream_t stream)
{
    (void)in_sizes; (void)n_in; (void)out_size; (void)ws_size;

    const float* in  = (const float*)d_in[0];
    float*       out = (float*)d_out;

    // Workspace layout: mask bits (8 MiB) | rowCounts (128 KiB) | rowOffsets (128 KiB)
    unsigned* maskWords  = (unsigned*)d_ws;
    unsigned* rowCounts  = maskWords + (size_t)BATCH * Hh * (Ww / 32);
    unsigned* rowOffsets = rowCounts + (size_t)BATCH * Hh;

    nms_mask_kernel<<<BATCH * Hh, 256, 0, stream>>>(in, maskWords, rowCounts);
    scan_kernel<<<BATCH, 256, 0, stream>>>(rowCounts, rowOffsets);

    const int total4 = BATCH * 3 * (KCAP / 4);          // float4 elements
    fill_kernel<<<total4 / 256, 256, 0, stream>>>(out);
    scatter_kernel<<<BATCH * Hh, 64, 0, stream>>>(maskWords, rowOffsets, out);
}